// ProjectionMatcher_12610023981100
// MI455X (gfx1250) — compile-verified
//
#include <hip/hip_runtime.h>
#include <math.h>

// ---------------------------------------------------------------------------
// ProjectionMatcher for MI455X (gfx1250, wave32).
// D=128, B=8, K=512. Output: maxcorr (B*K floats) then shifts (B*K*2 floats).
// Heavy stage = 4096 windowed cross-correlations done as f32 WMMA GEMMs.
// ---------------------------------------------------------------------------

#define DD 128
#define NB 8
#define NK 512
#define HX 65          // rfft half-width
#define KPAD 68        // HX padded to multiple of 4 for WMMA K-steps
#define WIN 32         // argmax window size (indices 48..79)
#define H0 48

typedef float v2f __attribute__((ext_vector_type(2)));
typedef float v8f __attribute__((ext_vector_type(8)));

__device__ __forceinline__ v8f wmma4(v2f a, v2f b, v8f c) {
  // V_WMMA_F32_16X16X4_F32 : D = A(16x4) * B(4x16) + C(16x16)
  return __builtin_amdgcn_wmma_f32_16x16x4_f32(
      /*neg_a=*/false, a, /*neg_b=*/false, b,
      /*c_mod=*/(short)0, c, /*reuse_a=*/false, /*reuse_b=*/false);
}

// ---------------------------------------------------------------------------
// Stockham radix-2 forward FFT, N=128, 64 threads, ping-pong LDS buffers.
// Returns pointer to buffer holding the natural-order result.
// ---------------------------------------------------------------------------
__device__ float2* fft128_fwd(float2* s0, float2* s1, int t) {
  float2* src = s0;
  float2* dst = s1;
#pragma unroll
  for (int st = 0; st < 7; ++st) {
    __syncthreads();
    int Ns = 1 << st;
    int r  = t & (Ns - 1);
    int q  = t >> st;
    float2 v0 = src[t];
    float2 v1 = src[t + 64];
    float ang = -3.14159265358979323846f * (float)r / (float)Ns;
    float c = cosf(ang), sn = sinf(ang);
    float2 w1 = make_float2(v1.x * c - v1.y * sn, v1.x * sn + v1.y * c);
    int d = (q << (st + 1)) + r;
    dst[d]      = make_float2(v0.x + w1.x, v0.y + w1.y);
    dst[d + Ns] = make_float2(v0.x - w1.x, v0.y - w1.y);
    float2* tmp = src; src = dst; dst = tmp;
  }
  __syncthreads();
  return src;
}

// ---- Volume rfftn (plain, unshifted; shifts folded in later) --------------
__global__ void k_fft_x_vol(const float* __restrict__ vol, float2* __restrict__ V) {
  __shared__ float2 sb[2][128];
  int row = blockIdx.x;            // z*128 + y
  int z = row >> 7, y = row & 127;
  int t = threadIdx.x;
  const float* p = vol + ((size_t)(z * 128 + y)) * 128;
  sb[0][t]      = make_float2(p[t], 0.0f);
  sb[0][t + 64] = make_float2(p[t + 64], 0.0f);
  float2* res = fft128_fwd(sb[0], sb[1], t);
  float2* o = V + ((size_t)(z * 128 + y)) * HX;
  o[t] = res[t];
  if (t == 0) o[64] = res[64];
}

__global__ void k_fft_y_vol(float2* __restrict__ V) {
  __shared__ float2 sb[2][128];
  int col = blockIdx.x;            // z*65 + kx
  int z = col / HX, kx = col - z * HX;
  int t = threadIdx.x;
  float2* base = V + ((size_t)z * 128) * HX + kx;
  sb[0][t]      = base[(size_t)t * HX];
  sb[0][t + 64] = base[(size_t)(t + 64) * HX];
  float2* res = fft128_fwd(sb[0], sb[1], t);
  base[(size_t)t * HX]        = res[t];
  base[(size_t)(t + 64) * HX] = res[t + 64];
}

__global__ void k_fft_z_vol(float2* __restrict__ V) {
  __shared__ float2 sb[2][128];
  int col = blockIdx.x;            // ky*65 + kx
  int ky = col / HX, kx = col - ky * HX;
  int t = threadIdx.x;
  float2* base = V + (size_t)ky * HX + kx;
  const size_t zs = (size_t)128 * HX;
  sb[0][t]      = base[(size_t)t * zs];
  sb[0][t + 64] = base[(size_t)(t + 64) * zs];
  float2* res = fft128_fwd(sb[0], sb[1], t);
  base[(size_t)t * zs]        = res[t];
  base[(size_t)(t + 64) * zs] = res[t + 64];
}

// ---- Particle rfft2 (plain) -----------------------------------------------
__global__ void k_fft_x_part(const float* __restrict__ parts, float2* __restrict__ RF) {
  __shared__ float2 sb[2][128];
  int row = blockIdx.x;            // b*128 + y
  int b = row >> 7, y = row & 127;
  int t = threadIdx.x;
  const float* p = parts + ((size_t)(b * 128 + y)) * 128;
  sb[0][t]      = make_float2(p[t], 0.0f);
  sb[0][t + 64] = make_float2(p[t + 64], 0.0f);
  float2* res = fft128_fwd(sb[0], sb[1], t);
  float2* o = RF + ((size_t)(b * 128 + y)) * HX;
  o[t] = res[t];
  if (t == 0) o[64] = res[64];
}

__global__ void k_fft_y_part(float2* __restrict__ RF) {
  __shared__ float2 sb[2][128];
  int col = blockIdx.x;            // b*65 + kx
  int b = col / HX, kx = col - b * HX;
  int t = threadIdx.x;
  float2* base = RF + ((size_t)b * 128) * HX + kx;
  sb[0][t]      = base[(size_t)t * HX];
  sb[0][t + 64] = base[(size_t)(t + 64) * HX];
  float2* res = fft128_fwd(sb[0], sb[1], t);
  base[(size_t)t * HX]        = res[t];
  base[(size_t)(t + 64) * HX] = res[t + 64];
}

// ---- Inverse-DFT twiddle tables (window-restricted) -----------------------
// Ex[kx][xw]  = e^{+2*pi*i * kx * xpre / 128},  xpre = (xw+48+64) & 127
// EyT[yw][ky] = e^{+2*pi*i * ky * ypre / 128},  EyTmi stores -Im part.
__global__ void k_tables(float* __restrict__ Exr, float* __restrict__ Exi,
                         float* __restrict__ EyTr, float* __restrict__ EyTmi) {
  const float twopi_over = 6.283185307179586f / 128.0f;
  int tid = threadIdx.x;
  for (int i = tid; i < KPAD * WIN; i += blockDim.x) {
    int kx = i / WIN, xw = i - (i / WIN) * WIN;
    float vr = 0.0f, vi = 0.0f;
    if (kx < HX) {
      int xpre = (xw + 112) & 127;
      int m = (kx * xpre) & 127;
      float ang = twopi_over * (float)m;
      vr = cosf(ang); vi = sinf(ang);
    }
    Exr[i] = vr; Exi[i] = vi;
  }
  for (int i = tid; i < WIN * 128; i += blockDim.x) {
    int yw = i / 128, ky = i - (i / 128) * 128;
    int ypre = (yw + 112) & 127;
    int m = (ky * ypre) & 127;
    float ang = twopi_over * (float)m;
    EyTr[i]  = cosf(ang);
    EyTmi[i] = -sinf(ang);
  }
}

// ---- Slice extraction: Pc[k,ky,kx] ----------------------------------------
// Pc = conj(projs[k, (ky+64)&127, kx]) * (-1)^{ky+kx} * c_kx / D^2
// vol_dft_shifted[zs,ys,x] = V[(zs+64)&127,(ys+64)&127,x] * (-1)^{zs+ys+x}
__global__ void k_extract(const float* __restrict__ rot,
                          const float2* __restrict__ V,
                          float2* __restrict__ Pc) {
  int k = blockIdx.y;
  int idx = blockIdx.x * blockDim.x + threadIdx.x;
  if (idx >= 128 * HX) return;
  int ky = idx / HX, kx = idx - ky * HX;
  int ys = (ky + 64) & 127;

  float fy = (float)(ys - 64) * (1.0f / 128.0f);
  float fx = (float)kx * (1.0f / 128.0f);

  const float* R = rot + (size_t)k * 9;
  float X = R[0] * fx + R[1] * fy;
  float Y = R[3] * fx + R[4] * fy;
  float Z = R[6] * fx + R[7] * fy;

  bool flip = X < 0.0f;
  float s = flip ? -1.0f : 1.0f;
  float zi = Z * s * 128.0f + 64.0f;
  float yi = Y * s * 128.0f + 64.0f;
  float xi = X * s * 128.0f;

  float z0f = floorf(zi), y0f = floorf(yi), x0f = floorf(xi);
  float wz = zi - z0f, wy = yi - y0f, wx = xi - x0f;
  int z0 = (int)z0f, y0 = (int)y0f, x0 = (int)x0f;

  float vr = 0.0f, viq = 0.0f;
#pragma unroll
  for (int dz = 0; dz < 2; ++dz)
#pragma unroll
    for (int dy = 0; dy < 2; ++dy)
#pragma unroll
      for (int dx = 0; dx < 2; ++dx) {
        int zz = z0 + dz, yy = y0 + dy, xx = x0 + dx;
        bool valid = (zz >= 0) & (zz < 128) & (yy >= 0) & (yy < 128) &
                     (xx >= 0) & (xx < HX);
        float w = (dz ? wz : 1.0f - wz) * (dy ? wy : 1.0f - wy) *
                  (dx ? wx : 1.0f - wx);
        if (valid) {
          int zu = (zz + 64) & 127, yu = (yy + 64) & 127;
          float2 v = V[((size_t)zu * 128 + yu) * HX + xx];
          float sg = ((zz + yy + xx) & 1) ? -w : w;
          vr  += v.x * sg;
          viq += v.y * sg;
        }
      }
  if (flip) viq = -viq;
  bool inband = sqrtf(fx * fx + fy * fy) <= 0.5f;
  if (!inband) { vr = 0.0f; viq = 0.0f; }

  float cw = (kx == 0 || kx == 64) ? 1.0f : 2.0f;
  float sc = (((ky + kx) & 1) ? -cw : cw) * (1.0f / 16384.0f);
  Pc[((size_t)k * 128 + ky) * HX + kx] = make_float2(vr * sc, -viq * sc);
}

// ---- Correlation: per (b,k) pair, WMMA GEMMs + windowed argmax ------------
// LDS (dynamic, floats): Qr[128*68] Qi[128*68] Tr[128*32] Ti[128*32] = 100 KB
__global__ void k_corr(const float2* __restrict__ RF, const float2* __restrict__ Pc,
                       const float* __restrict__ Exr, const float* __restrict__ Exi,
                       const float* __restrict__ EyTr, const float* __restrict__ EyTmi,
                       float* __restrict__ out) {
  extern __shared__ float smem[];
  float* Qr = smem;                     // 8704 floats
  float* Qi = smem + 8704;              // 8704 floats
  float* Tr = smem + 17408;             // 4096 floats
  float* Ti = smem + 21504;             // 4096 floats
  float* corrLds = smem;                // reuse Qr after stage 1 (1024 floats)
  float* redv = Qi;                     // reuse Qi during reduction
  int*   redi = (int*)(Qi + 256);

  const int k = blockIdx.x;
  const int b = blockIdx.y;
  const int tid  = threadIdx.x;
  const int wave = tid >> 5;
  const int lane = tid & 31;
  const int l15  = lane & 15;           // M row (A) / N col (B,C,D)
  const int kb   = (lane >> 4) << 1;    // K sub-offset within 4-wide step
  const int half8 = (lane >> 4) << 3;   // +8 rows for upper half in C/D

  // Phase A: Q = RF[b] * Pc[k] elementwise (complex), zero-padded to KPAD.
  {
    const float2* rfb = RF + (size_t)b * 128 * HX;
    const float2* pck = Pc + (size_t)k * 128 * HX;
    for (int i = tid; i < 128 * KPAD; i += 256) {
      int ky = i / KPAD, kx = i - ky * KPAD;
      float qr = 0.0f, qi = 0.0f;
      if (kx < HX) {
        float2 a = rfb[(size_t)ky * HX + kx];
        float2 p = pck[(size_t)ky * HX + kx];
        qr = a.x * p.x - a.y * p.y;
        qi = a.x * p.y + a.y * p.x;
      }
      Qr[i] = qr;
      Qi[i] = qi;
    }
  }
  __syncthreads();

  // Phase B: T(128x32 complex) = Q(128x68 complex) x Ex(68x32 complex)
  // 16 complex 16x16 tiles (8 M-tiles x 2 N-tiles), 2 per wave.
#pragma unroll
  for (int rep = 0; rep < 2; ++rep) {
    int job = wave + (rep << 3);        // 0..15
    int mt = job >> 1, nt = job & 1;
    int arow = (mt * 16 + l15) * KPAD;
    int ncol = nt * 16 + l15;
    v8f accr = {};
    v8f acci = {};
    for (int kk = 0; kk < KPAD / 4; ++kk) {
      int ka = (kk << 2) + kb;
      v2f ar = { Qr[arow + ka], Qr[arow + ka + 1] };
      v2f ai = { Qi[arow + ka], Qi[arow + ka + 1] };
      v2f br = { Exr[ka * WIN + ncol], Exr[(ka + 1) * WIN + ncol] };
      v2f bi = { Exi[ka * WIN + ncol], Exi[(ka + 1) * WIN + ncol] };
      accr = wmma4(ar, br, accr);
      accr = wmma4(-ai, bi, accr);      // Tr -= Qi*Exi (f32 WMMA: negate A in VALU)
      acci = wmma4(ar, bi, acci);
      acci = wmma4(ai, br, acci);
    }
    int mo = mt * 16 + half8;
#pragma unroll
    for (int j = 0; j < 8; ++j) {
      Tr[(mo + j) * WIN + ncol] = accr[j];
      Ti[(mo + j) * WIN + ncol] = acci[j];
    }
  }
  __syncthreads();

  // Phase C: corr(32x32) = EyTr(32x128)*Tr + EyTmi(32x128)*Ti (real part).
  if (wave < 4) {
    int yt = wave >> 1, xt = wave & 1;
    int arow = (yt * 16 + l15) * 128;
    int ncol = xt * 16 + l15;
    v8f acc = {};
    for (int kk = 0; kk < 32; ++kk) {
      int ka = (kk << 2) + kb;
      v2f aR  = { EyTr[arow + ka],  EyTr[arow + ka + 1] };
      v2f aMi = { EyTmi[arow + ka], EyTmi[arow + ka + 1] };
      v2f bR  = { Tr[ka * WIN + ncol], Tr[(ka + 1) * WIN + ncol] };
      v2f bI  = { Ti[ka * WIN + ncol], Ti[(ka + 1) * WIN + ncol] };
      acc = wmma4(aR, bR, acc);
      acc = wmma4(aMi, bI, acc);
    }
    int yo = yt * 16 + half8;
#pragma unroll
    for (int j = 0; j < 8; ++j)
      corrLds[(yo + j) * WIN + xt * 16 + l15] = acc[j];
  }
  __syncthreads();

  // Phase D: max + first-occurrence (row-major) argmax over 32x32 window.
  float bestv = -INFINITY;
  int besti = 0;
  int base = tid * 4;
#pragma unroll
  for (int i = 0; i < 4; ++i) {
    float v = corrLds[base + i];
    if (v > bestv) { bestv = v; besti = base + i; }
  }
  __syncthreads();               // corrLds reads done before redv (aliased Qi? no) writes
  redv[tid] = bestv;
  redi[tid] = besti;
  __syncthreads();
  for (int s = 128; s > 0; s >>= 1) {
    if (tid < s) {
      float v2 = redv[tid + s];
      int   i2 = redi[tid + s];
      if (v2 > redv[tid] || (v2 == redv[tid] && i2 < redi[tid])) {
        redv[tid] = v2;
        redi[tid] = i2;
      }
    }
    __syncthreads();
  }
  if (tid == 0) {
    float mv = redv[0];
    int li = redi[0];
    int pair = b * NK + k;
    out[pair] = mv;
    out[NB * NK + pair * 2 + 0] = (float)(H0 + (li & 31));   // sx
    out[NB * NK + pair * 2 + 1] = (float)(H0 + (li >> 5));   // sy
  }
}

// ---------------------------------------------------------------------------
extern "C" void kernel_launch(void* const* d_in, const int* in_sizes, int n_in,
                              void* d_out, int out_size, void* d_ws, size_t ws_size,
                              hipStream_t stream) {
  (void)in_sizes; (void)n_in; (void)out_size; (void)ws_size;
  const float* vol   = (const float*)d_in[0];   // (128,128,128) f32
  const float* parts = (const float*)d_in[1];   // (8,128,128)   f32
  const float* rot   = (const float*)d_in[2];   // (512,3,3)     f32
  float* out = (float*)d_out;

  char* ws = (char*)d_ws;
  size_t off = 0;
  auto take = [&](size_t bytes) -> size_t {
    size_t o = off;
    off = (off + bytes + 255) & ~(size_t)255;
    return o;
  };
  float2* V    = (float2*)(ws + take((size_t)128 * 128 * HX * sizeof(float2)));
  float2* Pc   = (float2*)(ws + take((size_t)NK * 128 * HX * sizeof(float2)));
  float2* RF   = (float2*)(ws + take((size_t)NB * 128 * HX * sizeof(float2)));
  float*  Exr  = (float*)(ws + take((size_t)KPAD * WIN * sizeof(float)));
  float*  Exi  = (float*)(ws + take((size_t)KPAD * WIN * sizeof(float)));
  float*  EyTr = (float*)(ws + take((size_t)WIN * 128 * sizeof(float)));
  float*  EyTmi= (float*)(ws + take((size_t)WIN * 128 * sizeof(float)));

  // Volume rfftn (x real pass, then complex y/z passes, in place).
  k_fft_x_vol<<<128 * 128, 64, 0, stream>>>(vol, V);
  k_fft_y_vol<<<128 * HX, 64, 0, stream>>>(V);
  k_fft_z_vol<<<128 * HX, 64, 0, stream>>>(V);

  // Particle rfft2.
  k_fft_x_part<<<NB * 128, 64, 0, stream>>>(parts, RF);
  k_fft_y_part<<<NB * HX, 64, 0, stream>>>(RF);

  // Twiddle tables for the window-restricted inverse DFT.
  k_tables<<<1, 256, 0, stream>>>(Exr, Exi, EyTr, EyTmi);

  // Central-slice extraction (folds conj, fftshift signs, irfft weights).
  dim3 ge((128 * HX + 255) / 256, NK);
  k_extract<<<ge, 256, 0, stream>>>(rot, V, Pc);

  // Correlation + argmax: one block per (b,k) pair, 100 KB dynamic LDS.
  dim3 gc(NK, NB);
  k_corr<<<gc, 256, 102400, stream>>>(RF, Pc, Exr, Exi, EyTr, EyTmi, out);
}